// DeformableConv2d_18064632447113
// MI455X (gfx1250) — compile-verified
//
#include <hip/hip_runtime.h>
#include <hip/hip_bf16.h>

typedef __attribute__((ext_vector_type(2))) float v2f;
typedef __attribute__((ext_vector_type(8))) float v8f;

#define HW 4096          // 64*64 plane
#define CIN 256
#define COUT 256
#define KTOT 2304        // 9*256

// ---------------------------------------------------------------------------
// Kernel 1: weight[o][c][kh][kw] -> wt2[(k>>1)*512 + n*2 + (k&1)]
// with k = kk*256 + c  (kk-major K ordering so each K-chunk has fixed tap)
// ---------------------------------------------------------------------------
__global__ void dcn_wtrans_kernel(const float* __restrict__ w,
                                  float* __restrict__ wt2) {
    int gid = blockIdx.x * 256 + threadIdx.x;   // 2304*256 total
    int k = gid >> 8;        // 0..2303
    int n = gid & 255;       // output channel
    int c = k & 255;
    int kk = k >> 8;
    wt2[(size_t)(k >> 1) * 512 + n * 2 + (k & 1)] =
        w[(size_t)n * KTOT + c * 9 + kk];
}

// ---------------------------------------------------------------------------
// Kernel 2: fused deformable sampling + f32 WMMA GEMM
// block = 256 threads (8 wave32), tile = 32 spatial x 256 out-channels
// ---------------------------------------------------------------------------
__global__ __launch_bounds__(256)
void dcn_main_kernel(const float* __restrict__ x,
                     const float* __restrict__ off,
                     const float* __restrict__ bias,
                     const float* __restrict__ wt2,
                     float* __restrict__ out) {
    __shared__ __attribute__((aligned(16))) float4 s_tw[9 * 32];
    __shared__ __attribute__((aligned(16))) int4   s_ti[9 * 32];
    __shared__ __attribute__((aligned(16))) float  sA[32 * 34];   // padded
    __shared__ __attribute__((aligned(16))) float  sO[256 * 33];  // padded

    const int tid = threadIdx.x;
    const int m0  = blockIdx.x << 5;        // base spatial index (b,ho,wo)
    const int b   = m0 >> 12;               // / 4096
    const int rem = m0 & 4095;
    const int ho  = rem >> 6;
    const int wo0 = rem & 63;               // 0 or 32 -> fixed ho for all 32 rows

    // ---- build bilinear corner table: one entry per (tap, row) -------------
    for (int e = tid; e < 9 * 32; e += 256) {
        int kk = e >> 5, r = e & 31;
        int wo = wo0 + r;
        size_t ob = (size_t)(b * 18 + kk * 2) * HW + ho * 64 + wo;
        float oy = off[ob];
        float ox = off[ob + HW];
        float py = (float)(ho - 1 + kk / 3) + oy;
        float px = (float)(wo - 1 + kk % 3) + ox;
        float y0f = floorf(py), x0f = floorf(px);
        float wy1 = py - y0f, wx1 = px - x0f;
        float wy0 = 1.f - wy1, wx0 = 1.f - wx1;
        int y0 = (int)y0f, x0 = (int)x0f;
        int y1 = y0 + 1, x1 = x0 + 1;
        float vy0 = (y0 >= 0 && y0 < 64) ? 1.f : 0.f;
        float vy1 = (y1 >= 0 && y1 < 64) ? 1.f : 0.f;
        float vx0 = (x0 >= 0 && x0 < 64) ? 1.f : 0.f;
        float vx1 = (x1 >= 0 && x1 < 64) ? 1.f : 0.f;
        int cy0 = min(max(y0, 0), 63), cy1 = min(max(y1, 0), 63);
        int cx0 = min(max(x0, 0), 63), cx1 = min(max(x1, 0), 63);
        s_tw[e] = make_float4(wy0 * wx0 * vy0 * vx0, wy0 * wx1 * vy0 * vx1,
                              wy1 * wx0 * vy1 * vx0, wy1 * wx1 * vy1 * vx1);
        s_ti[e] = make_int4(cy0 * 64 + cx0, cy0 * 64 + cx1,
                            cy1 * 64 + cx0, cy1 * 64 + cx1);
    }
    __syncthreads();

    // ---- per-thread roles --------------------------------------------------
    const int lane = tid & 31;
    const int ln   = lane & 15;       // N / M column within tile
    const int lh   = lane >> 4;       // lane-half
    const int wv   = tid >> 5;        // wave id 0..7
    const int msub = wv >> 2;         // spatial sub-tile (rows 0-15 / 16-31)
    const int nb   = (wv & 3) << 6;   // out-channel base: 0/64/128/192
    const int row_s = tid & 31;       // staging: spatial row
    const int ci    = tid >> 5;       // staging: channel group 0..7
    const float* xb = x + (size_t)b * CIN * HW;

    v8f acc[4];
#pragma unroll
    for (int j = 0; j < 4; ++j) acc[j] = (v8f)0.0f;

    // ---- main K loop: 9 taps x 8 chunks of 32 channels ---------------------
    for (int kk = 0; kk < 9; ++kk) {
        float4 tw = s_tw[kk * 32 + row_s];
        int4   ti = s_ti[kk * 32 + row_s];
        for (int c0 = 0; c0 < 256; c0 += 32) {
            __syncthreads();   // previous chunk's compute finished
            // stage sampled A chunk [32 rows][32 channels] into LDS
#pragma unroll
            for (int u = 0; u < 4; ++u) {
                int c = c0 + (ci << 2) + u;
                const float* pl = xb + (size_t)c * HW;
                float v = tw.x * pl[ti.x] + tw.y * pl[ti.y] +
                          tw.z * pl[ti.z] + tw.w * pl[ti.w];
                sA[row_s * 34 + (ci << 2) + u] = v;
            }
            __syncthreads();
            const int kb2 = (kk * 256 + c0) >> 1;   // K-pair base
#pragma unroll
            for (int s = 0; s < 8; ++s) {           // 8 x (K=4) steps
                v2f a = *(const v2f*)&sA[(msub * 16 + ln) * 34 + s * 4 + lh * 2];
                const size_t kp = (size_t)(kb2 + s * 2 + lh) * 512;
#pragma unroll
                for (int j = 0; j < 4; ++j) {       // 4 N-tiles, A reused
                    v2f bf = *(const v2f*)&wt2[kp + (size_t)((nb + j * 16 + ln) << 1)];
                    acc[j] = __builtin_amdgcn_wmma_f32_16x16x4_f32(
                        false, a, false, bf, (short)0, acc[j], false, false);
                }
            }
        }
    }

    // ---- epilogue: bias, LDS transpose, coalesced store --------------------
#pragma unroll
    for (int j = 0; j < 4; ++j) {
        int n = nb + j * 16 + ln;
        float bv = bias[n];
#pragma unroll
        for (int v = 0; v < 8; ++v) {
            int m = msub * 16 + v + lh * 8;   // C/D layout: M = v + 8*lane_half
            sO[n * 33 + m] = acc[j][v] + bv;
        }
    }
    __syncthreads();
    float* op = out + (size_t)b * COUT * HW + ho * 64 + wo0;
    const int mloc = tid & 31;
#pragma unroll
    for (int i = 0; i < 32; ++i) {
        int n = (tid >> 5) + i * 8;
        op[(size_t)n * HW + mloc] = sO[n * 33 + mloc];
    }
}

// ---------------------------------------------------------------------------
extern "C" void kernel_launch(void* const* d_in, const int* in_sizes, int n_in,
                              void* d_out, int out_size, void* d_ws, size_t ws_size,
                              hipStream_t stream) {
    (void)in_sizes; (void)n_in; (void)out_size; (void)ws_size;
    const float* x      = (const float*)d_in[0];
    const float* offset = (const float*)d_in[1];
    const float* weight = (const float*)d_in[2];
    const float* bias   = (const float*)d_in[3];
    float* out = (float*)d_out;
    float* wt2 = (float*)d_ws;   // 2304/2 * 512 floats = 2.25 MB

    dcn_wtrans_kernel<<<KTOT, 256, 0, stream>>>(weight, wt2);
    dcn_main_kernel<<<16384 / 32, 256, 0, stream>>>(x, offset, bias, wt2, out);
}